// StructureEncoder_38208029065782
// MI455X (gfx1250) — compile-verified
//
#include <hip/hip_runtime.h>
#include <math.h>

// ---- problem constants (fixed by the reference) ----
#define HID     256
#define HEADS   8
#define CPH     32          // channels per head = HID/HEADS
#define GNUM    64          // num_graphs
#define NEG_SLOPE 0.2f
#define LN_EPS  1e-5f
#define KMAX    256         // max GEMM K (proj: 128, layers: 256)

typedef __attribute__((ext_vector_type(2))) float v2f;
typedef __attribute__((ext_vector_type(8))) float v8f;
typedef __attribute__((ext_vector_type(4))) unsigned u32x4;
typedef __attribute__((ext_vector_type(8))) int     i32x8;

// ---- order-preserving uint encoding of float (for integer atomicMax) ----
__device__ __forceinline__ unsigned enc_f32(float f) {
  unsigned b = __float_as_uint(f);
  return (b & 0x80000000u) ? ~b : (b | 0x80000000u);
}
__device__ __forceinline__ float dec_f32(unsigned o) {
  unsigned b = (o & 0x80000000u) ? (o ^ 0x80000000u) : ~o;
  return __uint_as_float(b);
}

// ---------------------------------------------------------------------------
// Tensor Data Mover: DMA a K x 16 fp32 panel of W[K,Nc] (columns n0..n0+15)
// from global memory into LDS at `lds` (row-major [K][16]).
// D# per CDNA5 ISA (2D tensor, groups 2/3 = NULL).
// The LDS address is passed INTO the asm so the shared array escapes and the
// "memory" clobber forces the compiler to treat LDS as written by the DMA.
// ---------------------------------------------------------------------------
__device__ __forceinline__ void tdm_load_w_panel(float* lds, const float* gaddr,
                                                 int K, int Nc) {
  union { u32x4 v; unsigned u[4]; } g0;
  union { i32x8 v; unsigned u[8]; } g1;
  const unsigned long long ga = (unsigned long long)(uintptr_t)gaddr;
  // generic-cast low 32 bits == LDS byte offset (aperture maps addr[31:0])
  const unsigned lds_off = (unsigned)(uintptr_t)(void*)lds;
  // group0: [1:0]=count=1 | lds_addr | global_addr[56:0] | [127:126]=type=2
  g0.u[0] = 1u;
  g0.u[1] = lds_off;                               // lds_addr (bytes)
  g0.u[2] = (unsigned)(ga & 0xFFFFFFFFu);          // global_addr[31:0]
  g0.u[3] = (unsigned)((ga >> 32) & 0x01FFFFFFu) | 0x80000000u;  // addr[56:32] | type=2
  // group1: data_size=2 (4B), dims/strides
  g1.u[0] = 0x00020000u;                           // wg_mask=0, data_size=4B
  g1.u[1] = ((unsigned)Nc & 0xFFFFu) << 16;        // tensor_dim0[15:0]
  g1.u[2] = ((unsigned)Nc >> 16) | (((unsigned)K & 0xFFFFu) << 16); // td0_hi | td1_lo
  g1.u[3] = ((unsigned)K >> 16) | (16u << 16);     // td1_hi | tile_dim0=16
  g1.u[4] = (unsigned)K & 0xFFFFu;                 // tile_dim1=K, tile_dim2=0
  g1.u[5] = (unsigned)Nc;                          // tensor_dim0_stride[31:0]
  g1.u[6] = 0u;                                    // stride0_hi | stride1_lo
  g1.u[7] = 0u;
  asm volatile("tensor_load_to_lds %0, %1"
               :: "s"(g0.v), "s"(g1.v), "s"(lds_off) : "memory");
}

// ---------------------------------------------------------------------------
// GEMM: C[M,Nc] = A[M,K] @ W[K,Nc] + bias[Nc]    (fp32, WMMA 16x16x4)
// Block = 8 waves: one 16-column weight panel staged in LDS by the TDM,
// each wave computes one 16x16 tile (8 consecutive m-tiles per block).
// EXEC is all-ones for every wave inside the WMMA loop (uniform branch).
// ---------------------------------------------------------------------------
__global__ __launch_bounds__(256)
void gemm_bias_wmma(const float* __restrict__ A, const float* __restrict__ W,
                    const float* __restrict__ bias, float* __restrict__ C,
                    int M, int K, int Nc) {
  __shared__ float sW[KMAX * 16];     // 16 KB panel: sW[k][c] = W[k][n0+c]

  const int wv   = threadIdx.x >> 5;
  const int lane = threadIdx.x & 31;
  const int n0   = blockIdx.x << 4;

  if (wv == 0) {                       // wave 0 kicks the DMA and waits on it
    tdm_load_w_panel(sW, W + n0, K, Nc);
    __builtin_amdgcn_s_wait_tensorcnt((short)0);
  }
  __syncthreads();

  const int mtiles = (M + 15) >> 4;
  const int mt = blockIdx.y * 8 + wv;
  if (mt < mtiles) {                   // wave-uniform: EXEC stays full
    const int m0    = mt << 4;
    const int rsel  = lane & 15;
    const int khalf = (lane >> 4) << 1;     // 0 for lanes 0-15, 2 for 16-31

    int arow = m0 + rsel; if (arow >= M) arow = M - 1;  // clamp, keep EXEC full
    const float* __restrict__ Ap = A + (size_t)arow * K + khalf;
    const float* __restrict__ Bp = sW + khalf * 16 + rsel;

    v8f acc = {};
    for (int k = 0; k < K; k += 4) {
      v2f a, b;
      a.x = Ap[k];
      a.y = Ap[k + 1];
      b.x = Bp[k * 16];
      b.y = Bp[k * 16 + 16];
      acc = __builtin_amdgcn_wmma_f32_16x16x4_f32(
          /*neg_a=*/false, a, /*neg_b=*/false, b,
          /*c_mod=*/(short)0, acc, /*reuse_a=*/false, /*reuse_b=*/false);
    }

    const float bv = bias[n0 + rsel];
    const int rbase = m0 + ((lane >> 4) << 3);   // lanes 16-31: rows m0+8..15
#pragma unroll
    for (int r = 0; r < 8; ++r) {
      const int row = rbase + r;
      if (row < M) C[(size_t)row * Nc + n0 + rsel] = acc[r] + bv;
    }
  }
}

// ---------------------------------------------------------------------------
// Pass 1: per-edge attention logits + segment-max (wave per edge).
// lane c covers column 32*head + c, so head == register index i.
// ---------------------------------------------------------------------------
__global__ __launch_bounds__(256)
void edge_logits_kernel(const float* __restrict__ xl, const float* __restrict__ xr,
                        const int* __restrict__ src, const int* __restrict__ dst,
                        const float* __restrict__ att,  // [HEADS,CPH]
                        float* __restrict__ logits,     // [E,HEADS]
                        unsigned* __restrict__ mxenc,   // [N,HEADS]
                        int E) {
  const int e = blockIdx.x * (blockDim.x >> 5) + (threadIdx.x >> 5);
  const int lane = threadIdx.x & 31;
  if (e >= E) return;
  const int s = src[e], d = dst[e];
  const float* __restrict__ pl = xl + (size_t)s * HID;
  const float* __restrict__ pr = xr + (size_t)d * HID;

  float t[HEADS];
#pragma unroll
  for (int i = 0; i < HEADS; ++i) {
    const int col = i * CPH + lane;
    float v = pl[col] + pr[col];
    v = (v > 0.0f) ? v : NEG_SLOPE * v;          // leaky_relu
    t[i] = att[i * CPH + lane] * v;
  }
#pragma unroll
  for (int off = 16; off >= 1; off >>= 1) {
#pragma unroll
    for (int i = 0; i < HEADS; ++i) t[i] += __shfl_xor(t[i], off, 32);
  }
  if (lane < HEADS) {
    const float v = t[lane];
    logits[(size_t)e * HEADS + lane] = v;
    atomicMax(&mxenc[(size_t)d * HEADS + lane], enc_f32(v));
  }
}

// ---------------------------------------------------------------------------
// Pass 2: ex = exp(logit - max[dst]); den[dst] += ex  (thread per edge*head)
// logits buffer is overwritten in-place with ex.
// ---------------------------------------------------------------------------
__global__ __launch_bounds__(256)
void edge_exp_kernel(float* __restrict__ logits_ex, const int* __restrict__ dst,
                     const unsigned* __restrict__ mxenc, float* __restrict__ den,
                     int EH) {
  const int i = blockIdx.x * blockDim.x + threadIdx.x;
  if (i >= EH) return;
  const int e = i >> 3;        // / HEADS
  const int h = i & (HEADS - 1);
  const int d = dst[e];
  const float m = dec_f32(mxenc[(size_t)d * HEADS + h]);
  const float ex = expf(logits_ex[i] - m);
  logits_ex[i] = ex;
  atomicAdd(&den[(size_t)d * HEADS + h], ex);
}

// ---------------------------------------------------------------------------
// Pass 3: out[dst] += alpha * xl[src]   (wave per edge, coalesced f32 atomics)
// ---------------------------------------------------------------------------
__global__ __launch_bounds__(256)
void edge_scatter_kernel(const float* __restrict__ xl, const float* __restrict__ ex,
                         const float* __restrict__ den, const int* __restrict__ src,
                         const int* __restrict__ dst, float* __restrict__ outacc,
                         int E) {
  const int e = blockIdx.x * (blockDim.x >> 5) + (threadIdx.x >> 5);
  const int lane = threadIdx.x & 31;
  if (e >= E) return;
  const int s = src[e], d = dst[e];
  const float* __restrict__ pl = xl + (size_t)s * HID;
  float* __restrict__ po = outacc + (size_t)d * HID;
#pragma unroll
  for (int i = 0; i < HEADS; ++i) {
    const float alpha = ex[(size_t)e * HEADS + i] /
                        fmaxf(den[(size_t)d * HEADS + i], 1e-16f);
    atomicAdd(&po[i * CPH + lane], alpha * pl[i * CPH + lane]);
  }
}

// ---------------------------------------------------------------------------
// Node epilogue: h = LayerNorm(h + elu(conv + conv_b))  (wave per node)
// ---------------------------------------------------------------------------
__global__ __launch_bounds__(256)
void node_update_kernel(float* __restrict__ h, const float* __restrict__ outacc,
                        const float* __restrict__ conv_b,
                        const float* __restrict__ ln_g, const float* __restrict__ ln_b,
                        int Nn) {
  const int n = blockIdx.x * (blockDim.x >> 5) + (threadIdx.x >> 5);
  const int lane = threadIdx.x & 31;
  if (n >= Nn) return;
  float* __restrict__ ph = h + (size_t)n * HID;
  const float* __restrict__ po = outacc + (size_t)n * HID;

  float y[HEADS];
  float sum = 0.0f;
#pragma unroll
  for (int i = 0; i < HEADS; ++i) {
    const int col = i * CPH + lane;
    const float conv = po[col] + conv_b[col];
    const float act = (conv > 0.0f) ? conv : (expf(conv) - 1.0f);   // elu
    y[i] = ph[col] + act;
    sum += y[i];
  }
#pragma unroll
  for (int off = 16; off >= 1; off >>= 1) sum += __shfl_xor(sum, off, 32);
  const float mu = sum * (1.0f / (float)HID);

  float var = 0.0f;
#pragma unroll
  for (int i = 0; i < HEADS; ++i) { const float dv = y[i] - mu; var += dv * dv; }
#pragma unroll
  for (int off = 16; off >= 1; off >>= 1) var += __shfl_xor(var, off, 32);
  var *= (1.0f / (float)HID);
  const float inv = rsqrtf(var + LN_EPS);

#pragma unroll
  for (int i = 0; i < HEADS; ++i) {
    const int col = i * CPH + lane;
    ph[col] = (y[i] - mu) * inv * ln_g[col] + ln_b[col];
  }
}

// ---------------------------------------------------------------------------
// Pooling
// ---------------------------------------------------------------------------
__global__ __launch_bounds__(256)
void pool_acc_kernel(const float* __restrict__ h, const int* __restrict__ batch,
                     float* __restrict__ sums, float* __restrict__ cnt, int Nn) {
  const int n = blockIdx.x * (blockDim.x >> 5) + (threadIdx.x >> 5);
  const int lane = threadIdx.x & 31;
  if (n >= Nn) return;
  const int g = batch[n];
  const float* __restrict__ ph = h + (size_t)n * HID;
#pragma unroll
  for (int i = 0; i < HEADS; ++i)
    atomicAdd(&sums[(size_t)g * HID + i * CPH + lane], ph[i * CPH + lane]);
  if (lane == 0) atomicAdd(&cnt[g], 1.0f);
}

__global__ __launch_bounds__(256)
void pool_div_kernel(const float* __restrict__ sums, const float* __restrict__ cnt,
                     float* __restrict__ out, int n) {
  const int i = blockIdx.x * blockDim.x + threadIdx.x;
  if (i < n) out[i] = sums[i] / fmaxf(cnt[i >> 8], 1.0f);   // HID == 256
}

// ---------------------------------------------------------------------------
// Utilities
// ---------------------------------------------------------------------------
__global__ __launch_bounds__(256)
void zero_kernel(float* __restrict__ p, size_t n) {
  const size_t i = (size_t)blockIdx.x * blockDim.x + threadIdx.x;
  if (i < n) p[i] = 0.0f;
}
__global__ __launch_bounds__(256)
void copy_f_kernel(const float* __restrict__ s, float* __restrict__ d, size_t n) {
  const size_t i = (size_t)blockIdx.x * blockDim.x + threadIdx.x;
  if (i < n) d[i] = s[i];
}
__global__ __launch_bounds__(256)
void copy_i_kernel(const int* __restrict__ s, int* __restrict__ d, size_t n) {
  const size_t i = (size_t)blockIdx.x * blockDim.x + threadIdx.x;
  if (i < n) d[i] = s[i];
}

// ---------------------------------------------------------------------------
// Host orchestration
// ---------------------------------------------------------------------------
extern "C" void kernel_launch(void* const* d_in, const int* in_sizes, int n_in,
                              void* d_out, int out_size, void* d_ws, size_t ws_size,
                              hipStream_t stream) {
  // inputs (setup_inputs order)
  const float* x       = (const float*)d_in[0];
  const int*   ei      = (const int*)  d_in[1];
  const int*   batch   = (const int*)  d_in[2];
  const float* proj_W  = (const float*)d_in[4];
  const float* proj_b  = (const float*)d_in[5];
  const float* Wl      = (const float*)d_in[6];
  const float* bl      = (const float*)d_in[7];
  const float* Wr      = (const float*)d_in[8];
  const float* br      = (const float*)d_in[9];
  const float* att     = (const float*)d_in[10];
  const float* conv_b  = (const float*)d_in[11];
  const float* ln_g    = (const float*)d_in[12];
  const float* ln_b    = (const float*)d_in[13];

  const int N  = in_sizes[2];           // 50000
  const int IN = in_sizes[0] / N;       // 128
  const int E  = in_sizes[1] / 2;
  const int L  = in_sizes[7] / HID;     // 5

  const int* src = ei;
  const int* dst = ei + E;

  // workspace carve (floats)
  float* ws = (float*)d_ws;
  size_t off = 0;
  float*    h      = ws + off; off += (size_t)N * HID;
  float*    xl     = ws + off; off += (size_t)N * HID;
  float*    xr     = ws + off; off += (size_t)N * HID;
  float*    outacc = ws + off; off += (size_t)N * HID;
  float*    exbuf  = ws + off; off += (size_t)E * HEADS;   // logits, then exp
  unsigned* mxenc  = (unsigned*)(ws + off); off += (size_t)N * HEADS;
  float*    den    = ws + off; off += (size_t)N * HEADS;
  float*    sums   = ws + off; off += (size_t)GNUM * HID;
  float*    cnt    = ws + off; off += (size_t)GNUM;
  (void)ws_size; (void)n_in; (void)out_size;

  float* out_emb = (float*)d_out;                       // [G,HID]
  float* out_h   = out_emb + (size_t)GNUM * HID;        // [N,HID]
  int*   out_bv  = (int*)d_out + (size_t)GNUM * HID + (size_t)N * HID;  // [N]

  const int TB = 256;
  auto cdiv = [](size_t a, size_t b) { return (int)((a + b - 1) / b); };

  const int mtiles = (N + 15) / 16;
  const dim3 ggrid(HID / 16, (unsigned)cdiv(mtiles, 8));

  // h = x @ proj_W + proj_b
  gemm_bias_wmma<<<ggrid, TB, 0, stream>>>(x, proj_W, proj_b, h, N, IN, HID);

  const int edgeWaveBlocks = cdiv(E, 8);
  const int nodeWaveBlocks = cdiv(N, 8);

  for (int l = 0; l < L; ++l) {
    // zero accumulators for this layer
    zero_kernel<<<cdiv((size_t)N * HID, TB), TB, 0, stream>>>(outacc, (size_t)N * HID);
    zero_kernel<<<cdiv((size_t)N * HEADS, TB), TB, 0, stream>>>((float*)mxenc, (size_t)N * HEADS);
    zero_kernel<<<cdiv((size_t)N * HEADS, TB), TB, 0, stream>>>(den, (size_t)N * HEADS);

    gemm_bias_wmma<<<ggrid, TB, 0, stream>>>(
        h, Wl + (size_t)l * HID * HID, bl + (size_t)l * HID, xl, N, HID, HID);
    gemm_bias_wmma<<<ggrid, TB, 0, stream>>>(
        h, Wr + (size_t)l * HID * HID, br + (size_t)l * HID, xr, N, HID, HID);

    edge_logits_kernel<<<edgeWaveBlocks, TB, 0, stream>>>(
        xl, xr, src, dst, att + (size_t)l * HEADS * CPH, exbuf, mxenc, E);
    edge_exp_kernel<<<cdiv((size_t)E * HEADS, TB), TB, 0, stream>>>(
        exbuf, dst, mxenc, den, E * HEADS);
    edge_scatter_kernel<<<edgeWaveBlocks, TB, 0, stream>>>(
        xl, exbuf, den, src, dst, outacc, E);

    node_update_kernel<<<nodeWaveBlocks, TB, 0, stream>>>(
        h, outacc, conv_b + (size_t)l * HID, ln_g + (size_t)l * HID,
        ln_b + (size_t)l * HID, N);
  }

  // global mean pool
  zero_kernel<<<cdiv((size_t)GNUM * HID + GNUM, TB), TB, 0, stream>>>(
      sums, (size_t)GNUM * HID + GNUM);   // sums and cnt are contiguous
  pool_acc_kernel<<<nodeWaveBlocks, TB, 0, stream>>>(h, batch, sums, cnt, N);
  pool_div_kernel<<<cdiv((size_t)GNUM * HID, TB), TB, 0, stream>>>(
      sums, cnt, out_emb, GNUM * HID);

  // outputs 2 & 3: h and batch_vec
  copy_f_kernel<<<cdiv((size_t)N * HID, TB), TB, 0, stream>>>(h, out_h, (size_t)N * HID);
  copy_i_kernel<<<cdiv((size_t)N, TB), TB, 0, stream>>>(batch, out_bv, (size_t)N);
}